// GCN_78640851189892
// MI455X (gfx1250) — compile-verified
//
#include <hip/hip_runtime.h>
#include <hip/hip_bf16.h>
#include <math.h>

#define NN   100000
#define EE   3200000
#define FIN  512
#define HID  16
#define NCLS 2
#define NGR  256

typedef __attribute__((ext_vector_type(2))) float v2f;
typedef __attribute__((ext_vector_type(8))) float v8f;

// ---------------- utility: zero the accumulator region ----------------
__global__ void zero_kernel(float* __restrict__ p, int count) {
    int i = blockIdx.x * blockDim.x + threadIdx.x;
    if (i < count) p[i] = 0.0f;
}

// ---------------- degree count: deg[d] += 1 per edge ----------------
__global__ void deg_kernel(const int* __restrict__ dst, float* __restrict__ deg) {
    int e = blockIdx.x * blockDim.x + threadIdx.x;
    if (e < EE) atomicAdd(&deg[dst[e]], 1.0f);
}

// ---------------- dinv = rsqrt(deg + 1)  (self loop) ----------------
__global__ void dinv_kernel(const float* __restrict__ deg, float* __restrict__ dinv) {
    int n = blockIdx.x * blockDim.x + threadIdx.x;
    if (n < NN) dinv[n] = rsqrtf(deg[n] + 1.0f);
}

// ---------------- h1 = x @ W1 via V_WMMA_F32_16X16X4_F32 ----------------
// One wave computes one 16x16 output tile (16 nodes x all 16 hidden chans).
// 6250 tiles exactly (N = 16*6250); K = 512 = 128 WMMA steps of K=4.
// A-frag layout (32-bit A 16x4): lane<16 -> m=lane, K offsets {0,1};
//                                lane>=16 -> m=lane-16, K offsets {2,3}.
// B-frag symmetric: lane holds W1[k+kb, n=lane&15], W1[k+kb+1, n].
// C/D layout: VGPR r holds row m = r + 8*(lane>=16), col n = lane&15.
__global__ __launch_bounds__(64) void gemm1_wmma(const float* __restrict__ x,
                                                 const float* __restrict__ W1,
                                                 float* __restrict__ h1) {
    int wave = (blockIdx.x * 64 + (int)threadIdx.x) >> 5;   // 0..6249, exact
    int lane = threadIdx.x & 31;
    int half = lane >> 4;            // 0 or 1
    int kb   = half << 1;            // 0 or 2
    int ncol = lane & 15;
    int m0   = wave * 16;
    int mrow = m0 + (lane & 15);

    const float* xrow = x  + (size_t)mrow * FIN + kb;   // 8B-aligned (kb even)
    const float* wcol = W1 + kb * HID + ncol;

    v8f c = {};
#pragma unroll 8
    for (int k = 0; k < FIN; k += 4) {
        v2f a = *(const v2f*)(xrow + k);                // global_load_b64
        v2f b;
        b.x = wcol[k * HID];
        b.y = wcol[k * HID + HID];
        // (neg_a, A, neg_b, B, c_mod, C, reuse_a, reuse_b)
        c = __builtin_amdgcn_wmma_f32_16x16x4_f32(false, a, false, b,
                                                  (short)0, c, false, false);
    }

    int mbase = m0 + (half << 3);
#pragma unroll
    for (int r = 0; r < 8; ++r)
        h1[(size_t)(mbase + r) * HID + ncol] = c[r];
}

// ---------------- edge scatter layer 1: agg1[d] += norm * h1[s] ----------------
// 4 lanes per edge, each handles 4 channels (float4 read, 4 f32 atomics).
__global__ void edge_agg1(const int* __restrict__ src, const int* __restrict__ dst,
                          const float* __restrict__ dinv, const float* __restrict__ h1,
                          float* __restrict__ agg1) {
    int t = blockIdx.x * blockDim.x + threadIdx.x;
    int e = t >> 2;
    if (e >= EE) return;
    int j0 = (t & 3) << 2;
    int s = src[e], d = dst[e];
    float nrm = dinv[s] * dinv[d];
    const float4 v = *(const float4*)(h1 + (size_t)s * HID + j0);
    float* o = agg1 + (size_t)d * HID + j0;
    atomicAdd(o + 0, nrm * v.x);
    atomicAdd(o + 1, nrm * v.y);
    atomicAdd(o + 2, nrm * v.z);
    atomicAdd(o + 3, nrm * v.w);
}

// ---------------- fused: self-loop + bias + ReLU + layer-2 linear ----------------
__global__ void node1_kernel(const float* __restrict__ agg1, const float* __restrict__ h1,
                             const float* __restrict__ dinv, const float* __restrict__ b1,
                             const float* __restrict__ W2, float* __restrict__ h2) {
    int n = blockIdx.x * blockDim.x + threadIdx.x;
    if (n >= NN) return;
    float di = dinv[n];
    float w  = di * di;                 // self-loop norm
    float a0 = 0.0f, a1 = 0.0f;
#pragma unroll
    for (int j = 0; j < HID; ++j) {
        float r = agg1[(size_t)n * HID + j] + w * h1[(size_t)n * HID + j] + b1[j];
        r = fmaxf(r, 0.0f);
        a0 = fmaf(r, W2[j * NCLS + 0], a0);
        a1 = fmaf(r, W2[j * NCLS + 1], a1);
    }
    h2[(size_t)n * NCLS + 0] = a0;
    h2[(size_t)n * NCLS + 1] = a1;
}

// ---------------- edge scatter layer 2 (2 channels per edge) ----------------
__global__ void edge_agg2(const int* __restrict__ src, const int* __restrict__ dst,
                          const float* __restrict__ dinv, const float* __restrict__ h2,
                          float* __restrict__ agg2) {
    int e = blockIdx.x * blockDim.x + threadIdx.x;
    if (e >= EE) return;
    int s = src[e], d = dst[e];
    float nrm = dinv[s] * dinv[d];
    atomicAdd(&agg2[(size_t)d * NCLS + 0], nrm * h2[(size_t)s * NCLS + 0]);
    atomicAdd(&agg2[(size_t)d * NCLS + 1], nrm * h2[(size_t)s * NCLS + 1]);
}

// ---------------- node finalize layer 2 + pooled sums ----------------
__global__ void node2_pool(const float* __restrict__ agg2, const float* __restrict__ h2,
                           const float* __restrict__ dinv, const float* __restrict__ b2,
                           const int* __restrict__ batch,
                           float* __restrict__ pool, float* __restrict__ cnt) {
    int n = blockIdx.x * blockDim.x + threadIdx.x;
    if (n >= NN) return;
    float di = dinv[n];
    float w  = di * di;
    float v0 = agg2[(size_t)n * NCLS + 0] + w * h2[(size_t)n * NCLS + 0] + b2[0];
    float v1 = agg2[(size_t)n * NCLS + 1] + w * h2[(size_t)n * NCLS + 1] + b2[1];
    int g = batch[n];
    atomicAdd(&pool[g * NCLS + 0], v0);
    atomicAdd(&pool[g * NCLS + 1], v1);
    atomicAdd(&cnt[g], 1.0f);
}

// ---------------- mean pool + log_softmax ----------------
__global__ void final_kernel(const float* __restrict__ pool, const float* __restrict__ cnt,
                             float* __restrict__ out) {
    int g = threadIdx.x;
    if (g >= NGR) return;
    float c  = fmaxf(cnt[g], 1.0f);
    float z0 = pool[g * NCLS + 0] / c;
    float z1 = pool[g * NCLS + 1] / c;
    float m  = fmaxf(z0, z1);
    float l  = logf(expf(z0 - m) + expf(z1 - m));
    out[g * NCLS + 0] = (z0 - m) - l;
    out[g * NCLS + 1] = (z1 - m) - l;
}

extern "C" void kernel_launch(void* const* d_in, const int* in_sizes, int n_in,
                              void* d_out, int out_size, void* d_ws, size_t ws_size,
                              hipStream_t stream) {
    const float* x   = (const float*)d_in[0];   // [N, 512]
    const float* W1  = (const float*)d_in[1];   // [512, 16]
    const float* b1  = (const float*)d_in[2];   // [16]
    const float* W2  = (const float*)d_in[3];   // [16, 2]
    const float* b2  = (const float*)d_in[4];   // [2]
    const int* eidx  = (const int*)d_in[5];     // [2, E]
    const int* batch = (const int*)d_in[6];     // [N]
    const int* src = eidx;
    const int* dst = eidx + EE;
    float* out = (float*)d_out;                 // [256, 2]

    // Workspace layout (floats)
    float* ws   = (float*)d_ws;
    float* h1   = ws;                           // 16N
    float* h2   = h1 + (size_t)16 * NN;         // 2N
    float* dinv = h2 + (size_t)2 * NN;          // N
    float* deg  = dinv + NN;                    // N   <- zeroed region begins here
    float* agg1 = deg + NN;                     // 16N
    float* agg2 = agg1 + (size_t)16 * NN;       // 2N
    float* pool = agg2 + (size_t)2 * NN;        // 512
    float* cnt  = pool + NGR * NCLS;            // 256
    int zcount  = 19 * NN + NGR * NCLS + NGR;   // deg+agg1+agg2+pool+cnt

    // 1) zero accumulators (must happen every call)
    zero_kernel<<<(zcount + 255) / 256, 256, 0, stream>>>(deg, zcount);
    // 2) in-degree count
    deg_kernel<<<(EE + 255) / 256, 256, 0, stream>>>(dst, deg);
    // 3) dinv = rsqrt(deg + 1)
    dinv_kernel<<<(NN + 255) / 256, 256, 0, stream>>>(deg, dinv);
    // 4) h1 = x @ W1  (WMMA, 6250 tiles, 2 waves/block)
    gemm1_wmma<<<(NN / 16) / 2, 64, 0, stream>>>(x, W1, h1);
    // 5) edge scatter layer 1
    edge_agg1<<<(4 * EE + 255) / 256, 256, 0, stream>>>(src, dst, dinv, h1, agg1);
    // 6) self-loop + bias + relu + layer-2 linear
    node1_kernel<<<(NN + 255) / 256, 256, 0, stream>>>(agg1, h1, dinv, b1, W2, h2);
    // 7) edge scatter layer 2
    edge_agg2<<<(EE + 255) / 256, 256, 0, stream>>>(src, dst, dinv, h2, agg2);
    // 8) finalize layer 2 + pooled atomics
    node2_pool<<<(NN + 255) / 256, 256, 0, stream>>>(agg2, h2, dinv, b2, batch, pool, cnt);
    // 9) mean + log_softmax -> d_out
    final_kernel<<<1, NGR, 0, stream>>>(pool, cnt, out);
}